// R_Sparse_Linear_14181982011638
// MI455X (gfx1250) — compile-verified
//
#include <hip/hip_runtime.h>
#include <hip/hip_bf16.h>
#include <cstdint>
#include <cstddef>

typedef float v2f __attribute__((ext_vector_type(2)));
typedef float v8f __attribute__((ext_vector_type(8)));

#define B_    4
#define DIN_  4096
#define DOUT_ 11008
#define RANK_ 512
#define KC_   512            // K-chunk size (DIN_ / 8; also == RANK_)
#define NCW_  8              // number of W K-chunks
#define NCH_  9              // total chunks: 8 over W + 1 over U

// ---------------------------------------------------------------------------
// Kernel 1: elementwise mask split.
//   xs = input where |input*scale| > th, else 0   (sparse path operand)
//   xl = input where |input*scale| <= th, else 0  (low-rank path operand)
// ---------------------------------------------------------------------------
__global__ void rsparse_mask_kernel(const float* __restrict__ input,
                                    const float* __restrict__ scale,
                                    const float* __restrict__ threshold,
                                    float* __restrict__ xs,
                                    float* __restrict__ xl) {
  int i = blockIdx.x * blockDim.x + threadIdx.x;
  if (i >= B_ * DIN_) return;
  int d = i & (DIN_ - 1);
  float x  = input[i];
  float th = threshold[0];
  bool m = fabsf(x * scale[d]) > th;
  xs[i] = m ? x : 0.0f;
  xl[i] = m ? 0.0f : x;
}

// ---------------------------------------------------------------------------
// Kernel 2a (split path): partial low-rank projection over a D-chunk.
//   lrpart[dc][b][r] = sum_{d in chunk dc} xl[b][d] * V[d][r]
// Grid (2, 8): blockIdx.x = 256-wide r block, blockIdx.y = d chunk.
// V reads perfectly coalesced across r; xl slice broadcast from LDS.
// ---------------------------------------------------------------------------
__global__ void __launch_bounds__(256) rsparse_lowrank_part_kernel(
    const float* __restrict__ xl,
    const float* __restrict__ V,
    float* __restrict__ lrpart) {   // [NCW_][B_][RANK_]
  __shared__ float xls[B_ * KC_];
  const int dc  = blockIdx.y;
  const int tid = threadIdx.x;
  for (int i = tid; i < B_ * KC_; i += 256) {
    int b = i / KC_, k = i & (KC_ - 1);
    xls[i] = xl[(size_t)b * DIN_ + dc * KC_ + k];
  }
  __syncthreads();

  const int r = blockIdx.x * 256 + tid;   // 0..511
  const float* Vp = V + (size_t)(dc * KC_) * RANK_ + r;
  float a0 = 0.f, a1 = 0.f, a2 = 0.f, a3 = 0.f;
  #pragma unroll 4
  for (int k = 0; k < KC_; ++k) {
    float v = Vp[(size_t)k * RANK_];
    a0 += xls[0 * KC_ + k] * v;
    a1 += xls[1 * KC_ + k] * v;
    a2 += xls[2 * KC_ + k] * v;
    a3 += xls[3 * KC_ + k] * v;
  }
  float* P = lrpart + (size_t)dc * (B_ * RANK_);
  P[0 * RANK_ + r] = a0;
  P[1 * RANK_ + r] = a1;
  P[2 * RANK_ + r] = a2;
  P[3 * RANK_ + r] = a3;
}

// Kernel 2b: reduce lrpart over chunks and apply S:  lr[b][r] = S[r]*sum_dc ...
__global__ void rsparse_lr_reduce_kernel(const float* __restrict__ lrpart,
                                         const float* __restrict__ S,
                                         float* __restrict__ lr) {
  int i = blockIdx.x * blockDim.x + threadIdx.x;  // 0..B_*RANK_-1
  if (i >= B_ * RANK_) return;
  int r = i & (RANK_ - 1);
  float s = 0.f;
  #pragma unroll
  for (int c = 0; c < NCW_; ++c) s += lrpart[(size_t)c * (B_ * RANK_) + i];
  lr[i] = s * S[r];
}

// ---------------------------------------------------------------------------
// Kernel 3 (split path): one (DOUT tile, K chunk) pair per wave.
// Transposed WMMA with V_WMMA_F32_16X16X4_F32:
//   C(16x16): M = 16 DOUT rows, N = batch (cols 0..3 valid)
//   A(16x4)  = W (or U) rows; lane L<16: K=kb..kb+1, lane L+16: K=2..3
//   B(4x16)  = xs^T (or lr^T) slice from LDS; lanes n>=4 read a clamped
//              in-bounds address -> garbage lands only in discarded columns.
// Grid (86, 9), 8 waves/block. Chunks 0..7: W K-slices; chunk 8: U path.
// Writes bias-free partials: partial[c][b][DOUT tile].
// ---------------------------------------------------------------------------
__global__ void __launch_bounds__(256) rsparse_gemm_chunk_kernel(
    const float* __restrict__ weight,
    const float* __restrict__ U,
    const float* __restrict__ xs,
    const float* __restrict__ lr,
    float* __restrict__ partial) {  // [NCH_][B_][DOUT_]
  __shared__ __align__(16) float b_lds[B_ * KC_];  // 8 KB

  const int  c   = blockIdx.y;        // 0..8
  const int  tid = threadIdx.x;
  const bool isU = (c == NCW_);

  const float* src       = isU ? lr : (xs + c * KC_);
  const int    srcStride = isU ? RANK_ : DIN_;
  for (int i = tid; i < B_ * KC_; i += 256) {
    int b = i / KC_, k = i & (KC_ - 1);
    b_lds[i] = src[(size_t)b * srcStride + k];
  }
  __syncthreads();

  const int lane = tid & 31;
  const int wave = tid >> 5;
  const int n    = lane & 15;          // output-tile column (batch index)
  const int kb   = (lane >> 4) << 1;   // per-lane-half K offset: 0 or 2
  const int o0   = (blockIdx.x * 8 + wave) * 16;
  const int row  = o0 + n;

  const float* A = isU ? (U + (size_t)row * RANK_ + kb)
                       : (weight + (size_t)row * DIN_ + (size_t)c * KC_ + kb);
  const int bbase = (n & 3) * KC_ + kb;

  v8f acc = {};
  #pragma unroll 16
  for (int k0 = 0; k0 < KC_; k0 += 4) {
    v2f a = *(const v2f*)(A + k0);                 // global_load_b64, streaming
    v2f b = *(const v2f*)(&b_lds[bbase + k0]);     // ds_load_b64
    acc = __builtin_amdgcn_wmma_f32_16x16x4_f32(
              false, a, false, b, (short)0, acc, false, false);
  }

  float* P = partial + (size_t)c * (B_ * DOUT_);
  const int mbase = (lane >> 4) * 8;   // C layout: VGPR v -> M=v / v+8
  if (n < B_) {
    #pragma unroll
    for (int v = 0; v < 8; ++v) {
      int m = o0 + mbase + v;
      P[(size_t)n * DOUT_ + m] = acc[v];
    }
  }
}

// Kernel 4: out[b][o] = bias[o] + sum_{c<9} partial[c][b][o]
__global__ void rsparse_out_reduce_kernel(const float* __restrict__ partial,
                                          const float* __restrict__ bias,
                                          float* __restrict__ out) {
  int i = blockIdx.x * blockDim.x + threadIdx.x;
  if (i >= B_ * DOUT_) return;
  int o = i % DOUT_;
  float s = bias[o];
  #pragma unroll
  for (int c = 0; c < NCH_; ++c) s += partial[(size_t)c * (B_ * DOUT_) + i];
  out[i] = s;
}

// ---------------------------------------------------------------------------
// Fallback fused kernels (used only if workspace is too small for partials).
// ---------------------------------------------------------------------------
__global__ void __launch_bounds__(256) rsparse_lowrank_fused_kernel(
    const float* __restrict__ xl, const float* __restrict__ V,
    const float* __restrict__ S, float* __restrict__ lr) {
  __shared__ float xls[B_ * DIN_];
  int tid = threadIdx.x;
  for (int i = tid; i < B_ * DIN_; i += blockDim.x) xls[i] = xl[i];
  __syncthreads();
  int r = blockIdx.x * blockDim.x + tid;
  float a0 = 0.f, a1 = 0.f, a2 = 0.f, a3 = 0.f;
  #pragma unroll 4
  for (int d = 0; d < DIN_; ++d) {
    float v = V[(size_t)d * RANK_ + r];
    a0 += xls[0 * DIN_ + d] * v;
    a1 += xls[1 * DIN_ + d] * v;
    a2 += xls[2 * DIN_ + d] * v;
    a3 += xls[3 * DIN_ + d] * v;
  }
  float s = S[r];
  lr[0 * RANK_ + r] = a0 * s;
  lr[1 * RANK_ + r] = a1 * s;
  lr[2 * RANK_ + r] = a2 * s;
  lr[3 * RANK_ + r] = a3 * s;
}

__global__ void __launch_bounds__(256) rsparse_main_fused_kernel(
    const float* __restrict__ weight, const float* __restrict__ bias,
    const float* __restrict__ U, const float* __restrict__ xs,
    const float* __restrict__ lr, float* __restrict__ out) {
  __shared__ __align__(16) float xs_lds[B_ * DIN_];
  __shared__ __align__(16) float lr_lds[B_ * RANK_];
  const int tid = threadIdx.x;
  for (int i = tid; i < B_ * DIN_;  i += 256) xs_lds[i] = xs[i];
  for (int i = tid; i < B_ * RANK_; i += 256) lr_lds[i] = lr[i];
  __syncthreads();
  const int lane = tid & 31;
  const int wave = tid >> 5;
  const int n    = lane & 15;
  const int kb   = (lane >> 4) << 1;
  const int o0   = (blockIdx.x * 8 + wave) * 16;
  const int row  = o0 + n;
  const float* Aw = weight + (size_t)row * DIN_  + kb;
  const float* Au = U      + (size_t)row * RANK_ + kb;
  const int bW = (n & 3) * DIN_  + kb;
  const int bU = (n & 3) * RANK_ + kb;
  v8f c = {};
  #pragma unroll 8
  for (int d0 = 0; d0 < DIN_; d0 += 4) {
    v2f a = *(const v2f*)(Aw + d0);
    v2f b = *(const v2f*)(&xs_lds[bW + d0]);
    c = __builtin_amdgcn_wmma_f32_16x16x4_f32(false, a, false, b, (short)0, c,
                                              false, false);
  }
  #pragma unroll 8
  for (int r0 = 0; r0 < RANK_; r0 += 4) {
    v2f a = *(const v2f*)(Au + r0);
    v2f b = *(const v2f*)(&lr_lds[bU + r0]);
    c = __builtin_amdgcn_wmma_f32_16x16x4_f32(false, a, false, b, (short)0, c,
                                              false, false);
  }
  const int mbase = (lane >> 4) * 8;
  if (n < B_) {
    #pragma unroll
    for (int v = 0; v < 8; ++v) {
      int m = o0 + mbase + v;
      out[(size_t)n * DOUT_ + m] = c[v] + bias[m];
    }
  }
}

// ---------------------------------------------------------------------------
extern "C" void kernel_launch(void* const* d_in, const int* in_sizes, int n_in,
                              void* d_out, int out_size, void* d_ws, size_t ws_size,
                              hipStream_t stream) {
  const float* input     = (const float*)d_in[0];  // [4,1,4096]
  const float* weight    = (const float*)d_in[1];  // [11008,4096]
  const float* bias      = (const float*)d_in[2];  // [11008]
  const float* U         = (const float*)d_in[3];  // [11008,512]
  const float* S         = (const float*)d_in[4];  // [512]
  const float* V         = (const float*)d_in[5];  // [4096,512]
  const float* scale     = (const float*)d_in[6];  // [4096]
  const float* threshold = (const float*)d_in[7];  // [1]
  float* out = (float*)d_out;                      // [4,1,11008]

  // workspace layout (floats)
  float* xs      = (float*)d_ws;                   // B_*DIN_
  float* xl      = xs + B_ * DIN_;                 // B_*DIN_
  float* lr      = xl + B_ * DIN_;                 // B_*RANK_
  float* lrpart  = lr + B_ * RANK_;                // NCW_*B_*RANK_
  float* partial = lrpart + NCW_ * B_ * RANK_;     // NCH_*B_*DOUT_
  const size_t needFloats =
      (size_t)(2 * B_ * DIN_) + (size_t)B_ * RANK_ +
      (size_t)NCW_ * B_ * RANK_ + (size_t)NCH_ * B_ * DOUT_;

  rsparse_mask_kernel<<<(B_ * DIN_ + 255) / 256, 256, 0, stream>>>(
      input, scale, threshold, xs, xl);

  if (ws_size >= needFloats * sizeof(float)) {
    // High-parallelism K-split path (deterministic partial-plane reduction).
    rsparse_lowrank_part_kernel<<<dim3(RANK_ / 256, NCW_), 256, 0, stream>>>(
        xl, V, lrpart);
    rsparse_lr_reduce_kernel<<<(B_ * RANK_ + 255) / 256, 256, 0, stream>>>(
        lrpart, S, lr);
    rsparse_gemm_chunk_kernel<<<dim3(DOUT_ / (16 * 8), NCH_), 256, 0, stream>>>(
        weight, U, xs, lr, partial);
    rsparse_out_reduce_kernel<<<(B_ * DOUT_ + 255) / 256, 256, 0, stream>>>(
        partial, bias, out);
  } else {
    // Fallback: fused single-pass kernels (smaller ws footprint).
    rsparse_lowrank_fused_kernel<<<RANK_ / 256, 256, 0, stream>>>(xl, V, S, lr);
    rsparse_main_fused_kernel<<<DOUT_ / (16 * 8), 256, 0, stream>>>(
        weight, bias, U, xs, lr, out);
  }
}